// IntraGNN_47210280517968
// MI455X (gfx1250) — compile-verified
//
#include <hip/hip_runtime.h>
#include <math.h>

#define EPSV  0.001f
#define SLOPE 0.2f
#define MB    32
#define NB    1024
#define DEDGE 64
#define DRAW  128
#define HIDN  128

typedef float v2f __attribute__((ext_vector_type(2)));
typedef float v8f __attribute__((ext_vector_type(8)));
typedef int   v4i __attribute__((ext_vector_type(4)));
typedef __attribute__((address_space(1))) v4i g_v4i;   // global
typedef __attribute__((address_space(3))) v4i l_v4i;   // LDS

#if __has_builtin(__builtin_amdgcn_global_load_async_to_lds_b128) && \
    __has_builtin(__builtin_amdgcn_s_wait_asynccnt)
#define USE_ASYNC_LDS 1
#else
#define USE_ASYNC_LDS 0
#endif

// ---------------- wave32 reductions ----------------
__device__ __forceinline__ int wsum_i(int v) {
#pragma unroll
  for (int o = 16; o > 0; o >>= 1) v += __shfl_xor(v, o, 32);
  return v;
}
__device__ __forceinline__ float wsum_f(float v) {
#pragma unroll
  for (int o = 16; o > 0; o >>= 1) v += __shfl_xor(v, o, 32);
  return v;
}
__device__ __forceinline__ float wmax_f(float v) {
#pragma unroll
  for (int o = 16; o > 0; o >>= 1) v = fmaxf(v, __shfl_xor(v, o, 32));
  return v;
}

// ---------------- kernel 1: squared norms of edge features ----------------
__global__ __launch_bounds__(256) void k_sq(const float* __restrict__ ef,
                                            float* __restrict__ sq) {
  int i = blockIdx.x * 256 + threadIdx.x;          // 0 .. M*N-1
  const float4* e4 = (const float4*)(ef + (size_t)i * DEDGE);
  float s = 0.f;
#pragma unroll
  for (int d = 0; d < DEDGE / 4; ++d) {
    float4 v = e4[d];
    s += v.x * v.x + v.y * v.y + v.z * v.z + v.w * v.w;
  }
  sq[i] = s;
}

// ---------------- kernel 2: per-row distance + exact top-k selection ----------------
// One wave (32 lanes) per row (m,j). Block = 4 waves.
__global__ __launch_bounds__(128) void k_select(const float* __restrict__ w,
                                                const float* __restrict__ ef,
                                                const float* __restrict__ sq,
                                                const float* __restrict__ thr_p,
                                                unsigned* __restrict__ sel,
                                                float* __restrict__ score_acc,
                                                int* __restrict__ total_acc) {
  __shared__ float eL[4][DEDGE];
  __shared__ float distS[4][NB];

  const int lane = threadIdx.x & 31;
  const int wv   = threadIdx.x >> 5;
  const int row  = blockIdx.x * 4 + wv;            // 0 .. M*N-1
  const int m    = row >> 10;
  const int j    = row & (NB - 1);

  // stage e_j into LDS (64 floats per wave)
  const float* ej = ef + ((size_t)m * NB + j) * DEDGE;
  eL[wv][lane]      = ej[lane];
  eL[wv][lane + 32] = ej[lane + 32];
  __syncthreads();

  const float thr  = thr_p[0];
  const float sqj  = sq[row];
  const float* wrow = w + (size_t)row * NB;

  unsigned maskbits = 0;
  for (int c = 0; c < 32; ++c) {
    const int k = c * 32 + lane;
    if (wrow[k] > EPSV) maskbits |= (1u << c);
    const float4* ek4 = (const float4*)(ef + ((size_t)m * NB + k) * DEDGE);
    float g = 0.f;
#pragma unroll
    for (int d = 0; d < DEDGE / 4; ++d) {
      float4 v = ek4[d];
      g += eL[wv][4 * d + 0] * v.x + eL[wv][4 * d + 1] * v.y +
           eL[wv][4 * d + 2] * v.z + eL[wv][4 * d + 3] * v.w;
    }
    float d2 = sqj + sq[m * NB + k] - 2.f * g;
    distS[wv][c * 32 + lane] = sqrtf(fmaxf(d2, 0.f));
  }

  const int counts = wsum_i(__popc(maskbits));
  const int ns_raw = (int)ceilf((float)counts * thr);
  const int ns     = ns_raw < counts ? ns_raw : counts;

  // masked max distance for normalization
  float mloc = -INFINITY;
  for (int c = 0; c < 32; ++c)
    if ((maskbits >> c) & 1) mloc = fmaxf(mloc, distS[wv][c * 32 + lane]);
  const float mx     = wmax_f(mloc);
  const float maxd   = (counts > 0) ? mx : 1.0f;
  const float invmax = (maxd > 0.f) ? 1.0f / maxd : 1.0f;
  const bool  single = (counts == 1);

  unsigned selword = 0;
  float    impsum  = 0.f;

  if (ns > 0) {                                    // wave-uniform branch
    // radix-select the ns-th smallest masked distance (float bits, monotone)
    unsigned T = 0u;
    for (int b = 31; b >= 0; --b) {
      const unsigned C = T | (1u << b);
      int cl = 0;
      for (int c = 0; c < 32; ++c)
        if ((maskbits >> c) & 1)
          cl += (__float_as_uint(distS[wv][c * 32 + lane]) < C);
      if (wsum_i(cl) < ns) T = C;
    }
    int clt = 0;
    for (int c = 0; c < 32; ++c)
      if ((maskbits >> c) & 1)
        clt += (__float_as_uint(distS[wv][c * 32 + lane]) < T);
    clt = wsum_i(clt);
    int r = ns - clt;                              // #ties (==T) to take, lowest index first

    const unsigned lmask = (lane == 0) ? 0u : (0xFFFFFFFFu >> (32 - lane));
    int prefix = 0;
    for (int c = 0; c < 32; ++c) {
      const bool  mk = (maskbits >> c) & 1;
      const float dv = distS[wv][c * 32 + lane];
      const unsigned db = __float_as_uint(dv);
      const bool lt = mk && (db < T);
      const bool eq = mk && (db == T);
      const unsigned balEq = (unsigned)__ballot(eq);
      const bool takeEq = eq && ((prefix + __popc(balEq & lmask)) < r);
      prefix += __popc(balEq);
      const bool s = lt || takeEq;
      const unsigned balSel = (unsigned)__ballot(s);
      if (lane == c) selword = balSel;
      if (s) impsum += single ? (1.0f - dv) : (1.0f - dv * invmax);
    }
  }

  sel[(size_t)row * 32 + lane] = selword;
  impsum = wsum_f(impsum);
  if (lane == 0) {
    atomicAdd(score_acc, impsum);
    atomicAdd(total_acc, ns_raw);
  }
}

// ------------- LDS staging helper (async-to-LDS when available) -------------
__device__ __forceinline__ void stage_chunk(float4* __restrict__ lbuf,
                                            const float4* __restrict__ gsrc,
                                            int tid, int n4_per_thread) {
#if USE_ASYNC_LDS
#pragma unroll
  for (int i = 0; i < 16; ++i) {
    if (i >= n4_per_thread) break;
    __builtin_amdgcn_global_load_async_to_lds_b128(
        (g_v4i*)(gsrc + tid + i * 256),
        (l_v4i*)(lbuf + tid + i * 256),
        0, 0);
  }
  __builtin_amdgcn_s_wait_asynccnt(0);
#else
  for (int i = 0; i < n4_per_thread; ++i)
    lbuf[tid + i * 256] = gsrc[tid + i * 256];
#endif
}

// ---------------- kernel 3: adjacency aggregation via WMMA f32 16x16x4 ----------------
// Block = 8 waves; each wave owns one 16-row j-tile; the 64x128 feature chunk
// is staged once per block into LDS and shared by all 8 waves.
__global__ __launch_bounds__(256) void k_agg(const float* __restrict__ feat,
                                             const unsigned* __restrict__ sel,
                                             float* __restrict__ tmp) {
  __shared__ float fbuf[64 * DRAW];                // 32 KB

  const int tid  = threadIdx.x;
  const int lane = tid & 31;
  const int wv   = tid >> 5;                       // 0..7
  const int m    = blockIdx.x >> 3;                // 0..31
  const int jt   = (blockIdx.x & 7) * 8 + wv;      // 0..63
  const int r    = lane & 15;                      // A-row / B-col / D-col
  const int hi   = lane >> 4;                      // K half select
  const int jglob = jt * 16 + r;

  const unsigned* srow = sel + ((size_t)m * NB + jglob) * 32;

  v8f acc[8];
#pragma unroll
  for (int t = 0; t < 8; ++t) acc[t] = (v8f){0.f, 0.f, 0.f, 0.f, 0.f, 0.f, 0.f, 0.f};

  for (int kc = 0; kc < NB / 64; ++kc) {
    const float4* gsrc =
        (const float4*)(feat + ((size_t)m * NB + (size_t)kc * 64) * DRAW);
    stage_chunk((float4*)fbuf, gsrc, tid, (64 * DRAW / 4) / 256);
    __syncthreads();

#pragma unroll
    for (int h = 0; h < 2; ++h) {
      const int wd = kc * 2 + h;
      unsigned word = srow[wd];
      if ((jglob >> 5) == wd) word |= 1u << (jglob & 31);   // min(sel + eye, 1)
#pragma unroll
      for (int q = 0; q < 8; ++q) {
        const int sh = q * 4 + hi * 2;
        v2f a;
        a.x = ((word >> sh) & 1u) ? 1.0f : 0.0f;
        a.y = ((word >> (sh + 1)) & 1u) ? 1.0f : 0.0f;
        const int kl = h * 32 + q * 4 + hi * 2;
        const float* fb = fbuf + kl * DRAW + r;
#pragma unroll
        for (int t = 0; t < 8; ++t) {
          v2f b;
          b.x = fb[t * 16];
          b.y = fb[DRAW + t * 16];
          acc[t] = __builtin_amdgcn_wmma_f32_16x16x4_f32(
              false, a, false, b, (short)0, acc[t], false, false);
        }
      }
    }
    __syncthreads();                               // protect fbuf before restage
  }

  float* trow = tmp + ((size_t)m * NB + (size_t)jt * 16) * DRAW;
#pragma unroll
  for (int t = 0; t < 8; ++t)
#pragma unroll
    for (int v = 0; v < 8; ++v) {
      const int Mr = v + hi * 8;
      trow[(size_t)Mr * DRAW + t * 16 + r] = acc[t][v];
    }
}

// ---------------- kernel 4: tmp @ w_gnn + leaky ReLU via WMMA f32 16x16x4 ----------------
// Block = 8 waves; w_gnn (64 KB) staged once into LDS, shared by all waves.
__global__ __launch_bounds__(256) void k_gemm2(const float* __restrict__ tmp,
                                               const float* __restrict__ wg,
                                               float* __restrict__ out) {
  __shared__ float wbuf[DRAW * HIDN];              // 64 KB

  const int tid  = threadIdx.x;
  const int lane = tid & 31;
  const int wv   = tid >> 5;
  const int rt   = blockIdx.x * 8 + wv;            // 0..2047 (16-row tiles)
  const int r    = lane & 15;
  const int hi   = lane >> 4;

  stage_chunk((float4*)wbuf, (const float4*)wg, tid, (DRAW * HIDN / 4) / 256);
  __syncthreads();

  v8f acc[8];
#pragma unroll
  for (int t = 0; t < 8; ++t) acc[t] = (v8f){0.f, 0.f, 0.f, 0.f, 0.f, 0.f, 0.f, 0.f};

  const float* arow = tmp + ((size_t)rt * 16 + r) * DRAW;
  for (int kc = 0; kc < 32; ++kc) {
    const int kb = kc * 4 + hi * 2;
    v2f a;
    a.x = arow[kb];
    a.y = arow[kb + 1];
    if (kc + 1 < 32) __builtin_prefetch(arow + kb + 4, 0, 0);
    const float* w0 = wbuf + kb * HIDN + r;
#pragma unroll
    for (int t = 0; t < 8; ++t) {
      v2f b;
      b.x = w0[t * 16];
      b.y = w0[HIDN + t * 16];
      acc[t] = __builtin_amdgcn_wmma_f32_16x16x4_f32(
          false, a, false, b, (short)0, acc[t], false, false);
    }
  }

#pragma unroll
  for (int t = 0; t < 8; ++t)
#pragma unroll
    for (int v = 0; v < 8; ++v) {
      float x = acc[t][v];
      x = (x >= 0.f) ? x : SLOPE * x;
      const int Mr = v + hi * 8;
      out[((size_t)rt * 16 + Mr) * HIDN + t * 16 + r] = x;
    }
}

// ---------------- kernel 5: finalize view_score ----------------
__global__ void k_final(const float* __restrict__ score_acc,
                        const int* __restrict__ total_acc,
                        float* __restrict__ out_score) {
  if (threadIdx.x == 0 && blockIdx.x == 0) {
    const float t = (float)(*total_acc);
    out_score[0] = (*score_acc) / t;
  }
}

// ---------------- launch ----------------
extern "C" void kernel_launch(void* const* d_in, const int* in_sizes, int n_in,
                              void* d_out, int out_size, void* d_ws, size_t ws_size,
                              hipStream_t stream) {
  const float* features = (const float*)d_in[0];   // [M,N,128]
  const float* weights  = (const float*)d_in[1];   // [M,N,N]
  const float* edge     = (const float*)d_in[2];   // [M,N,64]
  const float* thr      = (const float*)d_in[3];   // scalar
  // d_in[4] = batch_idx (arange, identity gather) -> unused
  const float* w_gnn    = (const float*)d_in[5];   // [128,128]
  float* out = (float*)d_out;                      // [M*N*128] out + [1] view_score

  char* ws = (char*)d_ws;
  float*    score_acc = (float*)(ws + 0);          // 4 B
  int*      total_acc = (int*)(ws + 8);            // 4 B
  float*    sq        = (float*)(ws + 1024);       // 128 KB
  unsigned* sel       = (unsigned*)(ws + (1u << 20));  // 4 MB at 1 MB
  float*    tmp       = (float*)(ws + (8u << 20));     // 16 MB at 8 MB

  (void)hipMemsetAsync(ws, 0, 16, stream);

  k_sq    <<<(MB * NB) / 256, 256, 0, stream>>>(edge, sq);
  k_select<<<(MB * NB) / 4,   128, 0, stream>>>(weights, edge, sq, thr, sel,
                                                score_acc, total_acc);
  k_agg   <<<MB * 8,          256, 0, stream>>>(features, sel, tmp);
  k_gemm2 <<<(MB * NB) / 128, 256, 0, stream>>>(tmp, w_gnn, out);
  k_final <<<1, 32, 0, stream>>>(score_acc, total_acc,
                                 out + (size_t)MB * NB * HIDN);
}